// HierarchyPrototypeClassifier_77120432767623
// MI455X (gfx1250) — compile-verified
//
#include <hip/hip_runtime.h>

#define B_N 8192
#define L_N 512
#define P_N 64
#define S_N 32
#define O_N 200
#define BIGF 1e30f

typedef __attribute__((ext_vector_type(2))) float v2f;
typedef __attribute__((ext_vector_type(8))) float v8f;

// ---------- helpers ----------
__device__ inline void atomicMinFloat(float* addr, float val) {
    unsigned int* ua = (unsigned int*)addr;
    unsigned int old = *ua;
    while (__uint_as_float(old) > val) {
        unsigned int assumed = old;
        old = atomicCAS(ua, assumed, __float_as_uint(val));
        if (old == assumed) break;
    }
}

// ---------- init scratch (must run every call; harness doesn't re-poison) ----------
__global__ void init_kernel(float* min1_sum, int* counts, float* grp_sum,
                            float* grp_min, float* col_min) {
    int t = blockIdx.x * blockDim.x + threadIdx.x;
    if (t == 0) min1_sum[0] = 0.0f;
    if (t < P_N) { counts[t] = 0; grp_sum[t] = 0.0f; col_min[t] = BIGF; }
    if (t < P_N * S_N) grp_min[t] = BIGF;
}

// ---------- row squared norms: one wave per row ----------
__global__ __launch_bounds__(32) void norms_kernel(const float* __restrict__ src,
                                                   float* __restrict__ dst, int len) {
    int row = blockIdx.x;
    int lane = threadIdx.x;
    const float* r = src + (size_t)row * len;
    float acc = 0.0f;
    for (int j = lane; j < len; j += 32) { float v = r[j]; acc += v * v; }
    for (int m = 16; m >= 1; m >>= 1) acc += __shfl_xor(acc, m, 32);
    if (lane == 0) dst[row] = acc;
}

// ---------- d_top = |x|^2 + |p|^2 - 2 x.p^T via V_WMMA_F32_16X16X4_F32 ----------
// one wave per 16(rows) x 64(all prototypes) slab: A fragment loaded once per
// K-step, reused across 4 N-tiles (4 accumulators). Grid: B/16 waves.
__global__ __launch_bounds__(32) void dtop_wmma_kernel(const float* __restrict__ x,
                                                       const float* __restrict__ proto,
                                                       const float* __restrict__ xsq,
                                                       const float* __restrict__ psq,
                                                       float* __restrict__ d_top) {
    int lane = threadIdx.x;
    int m0 = blockIdx.x * 16;
    int r = lane & 15;           // M for A-frag / N for B-frag
    int half = lane >> 4;        // 0 or 1
#if __has_builtin(__builtin_amdgcn_wmma_f32_16x16x4_f32)
    int kk = half << 1;          // K sub-offset: lanes 0-15 -> K{0,1}, 16-31 -> K{2,3}
    const float* xp = x + (size_t)(m0 + r) * L_N + kk;
    const float* pp = proto + (size_t)r * L_N + kk;   // N-tile t at byte offset t*16*L*4
    v8f acc0 = {}, acc1 = {}, acc2 = {}, acc3 = {};
    #pragma unroll 4
    for (int k = 0; k < L_N; k += 4) {
        v2f a, b0, b1, b2, b3;
        a.x  = xp[k];                 a.y  = xp[k + 1];
        b0.x = pp[k];                 b0.y = pp[k + 1];
        b1.x = pp[k + 16 * L_N];      b1.y = pp[k + 16 * L_N + 1];
        b2.x = pp[k + 32 * L_N];      b2.y = pp[k + 32 * L_N + 1];
        b3.x = pp[k + 48 * L_N];      b3.y = pp[k + 48 * L_N + 1];
        acc0 = __builtin_amdgcn_wmma_f32_16x16x4_f32(false, a, false, b0,
                                                     (short)0, acc0, false, false);
        acc1 = __builtin_amdgcn_wmma_f32_16x16x4_f32(false, a, false, b1,
                                                     (short)0, acc1, false, false);
        acc2 = __builtin_amdgcn_wmma_f32_16x16x4_f32(false, a, false, b2,
                                                     (short)0, acc2, false, false);
        acc3 = __builtin_amdgcn_wmma_f32_16x16x4_f32(false, a, false, b3,
                                                     (short)0, acc3, false, false);
    }
    #pragma unroll
    for (int g = 0; g < 8; ++g) {
        int m = m0 + g + 8 * half;               // C/D layout: VGPR g holds M=g (+8 upper half)
        float xm = xsq[m];
        float* drow = d_top + (size_t)m * P_N;
        drow[r]      = xm + psq[r]      - 2.0f * acc0[g];
        drow[r + 16] = xm + psq[r + 16] - 2.0f * acc1[g];
        drow[r + 32] = xm + psq[r + 32] - 2.0f * acc2[g];
        drow[r + 48] = xm + psq[r + 48] - 2.0f * acc3[g];
    }
#else
    // host-pass / fallback body: same result, scalar dots over all 64 columns
    for (int t = 0; t < 4; ++t) {
        int n = 16 * t + r;
        float pn = psq[n];
        for (int g = 0; g < 8; ++g) {
            int m = m0 + g + 8 * half;
            const float* xr = x + (size_t)m * L_N;
            const float* pr = proto + (size_t)n * L_N;
            float acc = 0.0f;
            for (int k = 0; k < L_N; ++k) acc += xr[k] * pr[k];
            d_top[(size_t)m * P_N + n] = xsq[m] + pn - 2.0f * acc;
        }
    }
#endif
}

// ---------- per-row min/argmin over P=64, accumulate min1, counts ----------
__global__ __launch_bounds__(32) void rowmin_kernel(const float* __restrict__ d_top,
                                                    int* __restrict__ idx,
                                                    int* __restrict__ counts,
                                                    float* __restrict__ min1_sum) {
    int b = blockIdx.x;
    int lane = threadIdx.x;
    const float* row = d_top + (size_t)b * P_N;
    float v0 = row[lane];
    float v1 = row[lane + 32];
    float best; int barg;
    if (v1 < v0) { best = v1; barg = lane + 32; } else { best = v0; barg = lane; }
    for (int m = 16; m >= 1; m >>= 1) {
        float ov = __shfl_xor(best, m, 32);
        int   oa = __shfl_xor(barg, m, 32);
        if (ov < best || (ov == best && oa < barg)) { best = ov; barg = oa; }
    }
    if (lane == 0) {
        idx[b] = barg;
        atomicAdd(min1_sum, best);
        atomicAdd(counts + barg, 1);
    }
}

// ---------- per-column min over B ----------
__global__ void colmin_kernel(const float* __restrict__ d_top, float* __restrict__ col_min) {
    __shared__ float sm[256];
    int p = blockIdx.x;
    float m = BIGF;
    for (int b = threadIdx.x; b < B_N; b += blockDim.x)
        m = fminf(m, d_top[(size_t)b * P_N + p]);
    sm[threadIdx.x] = m;
    __syncthreads();
    for (int s = 128; s >= 1; s >>= 1) {
        if ((int)threadIdx.x < s) sm[threadIdx.x] = fminf(sm[threadIdx.x], sm[threadIdx.x + s]);
        __syncthreads();
    }
    if (threadIdx.x == 0) col_min[p] = sm[0];
}

// ---------- gathered sub-distances d_sel[b,s], group mins and segment sums ----------
// one wave per sample; lane == s
__global__ __launch_bounds__(32) void dsel_kernel(const float* __restrict__ x,
                                                  const float* __restrict__ sub,
                                                  const float* __restrict__ xsq,
                                                  const float* __restrict__ ssq,
                                                  const int* __restrict__ idx,
                                                  float* __restrict__ d_sel,
                                                  float* __restrict__ grp_min,
                                                  float* __restrict__ grp_sum) {
    int b = blockIdx.x;
    int s = threadIdx.x;
    int g = idx[b];
    const float4* xr = (const float4*)(x + (size_t)b * L_N);
    const float4* sr = (const float4*)(sub + ((size_t)g * S_N + s) * L_N);
    float acc = 0.0f;
    #pragma unroll 4
    for (int j = 0; j < L_N / 4; ++j) {
        float4 a = xr[j], c = sr[j];
        acc += a.x * c.x + a.y * c.y + a.z * c.z + a.w * c.w;
    }
    float d = xsq[b] + ssq[g * S_N + s] - 2.0f * acc;
    d_sel[(size_t)b * S_N + s] = d;
    atomicMinFloat(grp_min + g * S_N + s, d);
    float m = d;
    for (int k = 16; k >= 1; k >>= 1) m = fminf(m, __shfl_xor(m, k, 32));
    if (s == 0) atomicAdd(grp_sum + g, m);
}

// ---------- per-group expert: out[b,o] = bias[g,o] + sum_s d_sel[b,s]*W[g,o,s] ----------
__global__ void out_kernel(const float* __restrict__ d_sel, const float* __restrict__ W,
                           const float* __restrict__ bias, const int* __restrict__ idx,
                           float* __restrict__ out) {
    int t = blockIdx.x * blockDim.x + threadIdx.x;
    if (t >= B_N * O_N) return;
    int b = t / O_N;
    int o = t - b * O_N;
    int g = idx[b];
    const float* ds = d_sel + (size_t)b * S_N;
    const float* w  = W + ((size_t)g * O_N + o) * S_N;
    float acc = bias[(size_t)g * O_N + o];
    #pragma unroll
    for (int s = 0; s < S_N; ++s) acc += ds[s] * w[s];
    out[t] = acc;
}

// ---------- final 4 scalars ----------
__global__ void finalize_kernel(const float* __restrict__ min1_sum,
                                const float* __restrict__ col_min,
                                const float* __restrict__ grp_min,
                                const float* __restrict__ grp_sum,
                                const int* __restrict__ counts,
                                float* __restrict__ out4) {
    __shared__ float s2[P_N], s3[P_N], s4[P_N];
    int p = threadIdx.x;  // blockDim == P_N == 64
    float gm = 0.0f;
    for (int s = 0; s < S_N; ++s) gm += grp_min[p * S_N + s];
    gm /= (float)S_N;
    int c = counts[p];
    s2[p] = col_min[p];
    s3[p] = (c > 0) ? gm : 0.0f;
    s4[p] = (c > 0) ? grp_sum[p] / (float)c : 0.0f;
    __syncthreads();
    for (int st = 32; st >= 1; st >>= 1) {
        if (p < st) { s2[p] += s2[p + st]; s3[p] += s3[p + st]; s4[p] += s4[p + st]; }
        __syncthreads();
    }
    if (p == 0) {
        out4[0] = min1_sum[0] / (float)B_N;  // min1
        out4[1] = s2[0] / (float)P_N;        // min2
        out4[2] = s3[0] / (float)P_N;        // sub_min1
        out4[3] = s4[0] / (float)P_N;        // sub_min2
    }
}

extern "C" void kernel_launch(void* const* d_in, const int* in_sizes, int n_in,
                              void* d_out, int out_size, void* d_ws, size_t ws_size,
                              hipStream_t stream) {
    const float* x     = (const float*)d_in[0];  // [B,L]
    const float* proto = (const float*)d_in[1];  // [P,L]
    const float* sub   = (const float*)d_in[2];  // [P,S,L]
    const float* W     = (const float*)d_in[3];  // [P,O,S]
    const float* bias  = (const float*)d_in[4];  // [P,O]

    float* out4 = (float*)d_out;                 // min1, min2, sub_min1, sub_min2
    float* out  = out4 + 4;                      // [B,O]

    // workspace layout (floats, then ints) ~3.2 MB
    float* ws       = (float*)d_ws;
    float* d_top    = ws;                        // B*P
    float* xsq      = d_top + (size_t)B_N * P_N; // B
    float* psq      = xsq + B_N;                 // P
    float* ssq      = psq + P_N;                 // P*S
    float* d_sel    = ssq + P_N * S_N;           // B*S
    float* col_min  = d_sel + (size_t)B_N * S_N; // P
    float* grp_min  = col_min + P_N;             // P*S
    float* grp_sum  = grp_min + P_N * S_N;       // P
    float* min1_sum = grp_sum + P_N;             // 1
    int*   idx      = (int*)(min1_sum + 1);      // B
    int*   counts   = idx + B_N;                 // P

    init_kernel<<<8, 256, 0, stream>>>(min1_sum, counts, grp_sum, grp_min, col_min);

    norms_kernel<<<B_N, 32, 0, stream>>>(x, xsq, L_N);
    norms_kernel<<<P_N, 32, 0, stream>>>(proto, psq, L_N);
    norms_kernel<<<P_N * S_N, 32, 0, stream>>>(sub, ssq, L_N);

    dtop_wmma_kernel<<<B_N / 16, 32, 0, stream>>>(x, proto, xsq, psq, d_top);

    rowmin_kernel<<<B_N, 32, 0, stream>>>(d_top, idx, counts, min1_sum);
    colmin_kernel<<<P_N, 256, 0, stream>>>(d_top, col_min);

    dsel_kernel<<<B_N, 32, 0, stream>>>(x, sub, xsq, ssq, idx, d_sel, grp_min, grp_sum);

    out_kernel<<<(B_N * O_N + 255) / 256, 256, 0, stream>>>(d_sel, W, bias, idx, out);

    finalize_kernel<<<1, P_N, 0, stream>>>(min1_sum, col_min, grp_min, grp_sum, counts, out4);
}